// CoxNNetLoss_14577119003171
// MI455X (gfx1250) — compile-verified
//
#include <hip/hip_runtime.h>
#include <hip/hip_bf16.h>

typedef __attribute__((ext_vector_type(2))) float v2f;
typedef __attribute__((ext_vector_type(8))) float v8f;

#define COX_N     16384
#define CHUNK     2048     // j-chunk staged in LDS: 2 arrays * 8KB = 16KB LDS
#define RED_BLKS  64

// ---------------------------------------------------------------------------
// Kernel 1: w[j] = exp(theta[j])
// ---------------------------------------------------------------------------
__global__ __launch_bounds__(256) void cox_exp_kernel(const float* __restrict__ theta,
                                                      float* __restrict__ w) {
    int i = blockIdx.x * 256 + threadIdx.x;
    if (i < COX_N) w[i] = expf(theta[i]);
}

// ---------------------------------------------------------------------------
// Kernel 2: risk[i] = sum_j (dur[j] >= dur[i]) * w[j]  via V_WMMA_F32_16X16X4_F32
//
// Block = 128 threads = 4 waves; wave t owns rows [blockIdx*64 + t*16, +16).
// A (16x4 f32): lane l -> row l%16; VGPR0/1 hold K = k0, k0+1 with
//               k0 = (l<16 ? 0 : 2)   (ISA 7.12.2, 32-bit A 16x4 layout).
// B (4x16 f32): row K broadcast across all 16 columns -> every column of D is
//               the same partial risk sum; mirrored K->half mapping as A.
// C/D (16x16 f32, 8 VGPRs): VGPR r of lane l = D[r + (l>=16 ? 8:0)][l%16].
// EXEC is all-ones (full blocks, no divergence) as WMMA requires.
// ---------------------------------------------------------------------------
__global__ __launch_bounds__(128) void cox_risk_wmma_kernel(const float* __restrict__ dur,
                                                            const float* __restrict__ w,
                                                            float* __restrict__ risk) {
    __shared__ __align__(16) float sDur[CHUNK];
    __shared__ __align__(16) float sW[CHUNK];

    const int tid  = threadIdx.x;
    const int wave = tid >> 5;          // wave32 on gfx1250
    const int lane = tid & 31;
    const int i0   = blockIdx.x * 64 + wave * 16;

    const float di = dur[i0 + (lane & 15)];       // this lane's row duration
    const int   k0 = (lane < 16) ? 0 : 2;         // K-pair owned by this half-wave

    v8f c0 = {};                                  // two accumulators to break the
    v8f c1 = {};                                  // WMMA D->C dependency chain

    for (int base = 0; base < COX_N; base += CHUNK) {
        __syncthreads();
        // Cooperative 128-bit staging: 2048 floats per array / 128 threads
        const float4* gd  = (const float4*)(dur + base);
        const float4* gw  = (const float4*)(w + base);
        float4*       s4d = (float4*)sDur;
        float4*       s4w = (float4*)sW;
        #pragma unroll
        for (int t = 0; t < (CHUNK / 4) / 128; ++t) {
            int idx = tid + t * 128;
            s4d[idx] = gd[idx];
            s4w[idx] = gw[idx];
        }
        __syncthreads();

        #pragma unroll 4
        for (int j = 0; j < CHUNK; j += 8) {
            // --- step A: K = j .. j+3 ---
            {
                const int jj = j + k0;                       // even -> 8B aligned
                v2f dj = *(const v2f*)(sDur + jj);           // ds_load_b64 (broadcast)
                v2f wj = *(const v2f*)(sW + jj);             // ds_load_b64 (broadcast)
                v2f a;
                a.x = (dj.x >= di) ? 1.0f : 0.0f;            // v_cmp + v_cndmask
                a.y = (dj.y >= di) ? 1.0f : 0.0f;
                c0 = __builtin_amdgcn_wmma_f32_16x16x4_f32(
                        false, a, false, wj, (short)0, c0, false, false);
            }
            // --- step B: K = j+4 .. j+7 (independent accumulator) ---
            {
                const int jj = j + 4 + k0;
                v2f dj = *(const v2f*)(sDur + jj);
                v2f wj = *(const v2f*)(sW + jj);
                v2f a;
                a.x = (dj.x >= di) ? 1.0f : 0.0f;
                a.y = (dj.y >= di) ? 1.0f : 0.0f;
                c1 = __builtin_amdgcn_wmma_f32_16x16x4_f32(
                        false, a, false, wj, (short)0, c1, false, false);
            }
        }
    }

    // All 16 columns of D are identical; lane 0 holds rows 0..7 in its 8 acc
    // VGPRs, lane 16 holds rows 8..15.
    v8f cc = c0 + c1;
    if (lane == 0) {
        #pragma unroll
        for (int r = 0; r < 8; ++r) risk[i0 + r] = cc[r];
    } else if (lane == 16) {
        #pragma unroll
        for (int r = 0; r < 8; ++r) risk[i0 + 8 + r] = cc[r];
    }
}

// ---------------------------------------------------------------------------
// Kernel 3: per-block partial sums of (theta - log(risk)) * events
// ---------------------------------------------------------------------------
__global__ __launch_bounds__(256) void cox_partial_kernel(const float* __restrict__ theta,
                                                          const float* __restrict__ ev,
                                                          const float* __restrict__ risk,
                                                          float* __restrict__ partials) {
    __shared__ float red[256];
    float acc = 0.0f;
    for (int i = blockIdx.x * 256 + threadIdx.x; i < COX_N; i += RED_BLKS * 256)
        acc += (theta[i] - logf(risk[i])) * ev[i];
    red[threadIdx.x] = acc;
    __syncthreads();
    #pragma unroll
    for (int s = 128; s > 0; s >>= 1) {
        if (threadIdx.x < s) red[threadIdx.x] += red[threadIdx.x + s];
        __syncthreads();
    }
    if (threadIdx.x == 0) partials[blockIdx.x] = red[0];
}

// ---------------------------------------------------------------------------
// Kernel 4: deterministic finalize -> out[0] = -sum / N
// ---------------------------------------------------------------------------
__global__ __launch_bounds__(64) void cox_finalize_kernel(const float* __restrict__ partials,
                                                          float* __restrict__ out) {
    __shared__ float red[RED_BLKS];
    red[threadIdx.x] = partials[threadIdx.x];
    __syncthreads();
    #pragma unroll
    for (int s = RED_BLKS / 2; s > 0; s >>= 1) {
        if (threadIdx.x < s) red[threadIdx.x] += red[threadIdx.x + s];
        __syncthreads();
    }
    if (threadIdx.x == 0) out[0] = -red[0] / (float)COX_N;
}

// ---------------------------------------------------------------------------
extern "C" void kernel_launch(void* const* d_in, const int* in_sizes, int n_in,
                              void* d_out, int out_size, void* d_ws, size_t ws_size,
                              hipStream_t stream) {
    const float* theta = (const float*)d_in[0];   // hazard_pred (N,1) flat
    const float* dur   = (const float*)d_in[1];   // durations (N)
    const float* ev    = (const float*)d_in[2];   // events (N)
    float*       out   = (float*)d_out;

    // workspace layout: w[N] | risk[N] | partials[RED_BLKS]
    float* w        = (float*)d_ws;
    float* risk     = w + COX_N;
    float* partials = risk + COX_N;

    cox_exp_kernel<<<COX_N / 256, 256, 0, stream>>>(theta, w);
    cox_risk_wmma_kernel<<<COX_N / 64, 128, 0, stream>>>(dur, w, risk);
    cox_partial_kernel<<<RED_BLKS, 256, 0, stream>>>(theta, ev, risk, partials);
    cox_finalize_kernel<<<1, RED_BLKS, 0, stream>>>(partials, out);
}